// MLPEdgePredictor_45887430591133
// MI455X (gfx1250) — compile-verified
//
#include <hip/hip_runtime.h>

typedef __attribute__((ext_vector_type(2))) float v2f;
typedef __attribute__((ext_vector_type(8))) float v8f;

#define D_DIM 128

// ---------------------------------------------------------------------------
// Phase 1: per 16-node tile, compute [p0,p1,q0,q1] per node with
// V_WMMA_F32_16X16X4_F32 (exact f32, matches reference numerics).
//   p_o = h[n] . W[o, 0:128],  q_o = h[n] . W[o, 128:256]
// B (4x16) column n uses W column pattern n&3; columns 4..15 are redundant
// replicas (never stored -- D columns are independent in D = A*B + C).
// ---------------------------------------------------------------------------
__global__ void __launch_bounds__(256)
edge_mlp_node_proj(const float* __restrict__ h,
                   const float* __restrict__ W,
                   float* __restrict__ p4,
                   int nNodes, int nTiles)
{
    const int lane = threadIdx.x & 31;
    const int wave = threadIdx.x >> 5;
    const int tile = blockIdx.x * (blockDim.x >> 5) + wave;
    if (tile >= nTiles) return;              // wave-uniform exit; EXEC stays all-1s

    const int m  = lane & 15;                // A row / B,D column within tile
    const int kh = lane >> 4;                // which K-pair this lane holds

    // A operand address: row of h (clamped; only matters for a partial tile)
    int row  = tile * 16 + m;
    int rowc = row < nNodes ? row : nNodes - 1;
    const float* hrow = h + (size_t)rowc * D_DIM;

    // B operand address: column n -> (o = n&1, half = n>>1), replicated mod 4
    const int n4 = m & 3;
    const float* wrow = W + (size_t)(n4 & 1) * (2 * D_DIM) + (size_t)(n4 >> 1) * D_DIM;

    v8f c = {0.f, 0.f, 0.f, 0.f, 0.f, 0.f, 0.f, 0.f};

    #pragma unroll 8
    for (int k = 0; k < D_DIM; k += 4) {
        const int kk = k + kh * 2;           // lanes 0-15: k,k+1 ; lanes 16-31: k+2,k+3
        v2f a = *(const v2f*)(hrow + kk);    // 8B aligned (kk even, row 512B aligned)
        v2f b = *(const v2f*)(wrow + kk);
        c = __builtin_amdgcn_wmma_f32_16x16x4_f32(
                /*neg_a=*/false, a, /*neg_b=*/false, b,
                /*c_mod=*/(short)0, c, /*reuse_a=*/false, /*reuse_b=*/false);
    }

    // D layout (16x16 f32): VGPR j, lanes 0-15 -> M=j, N=lane ; lanes 16-31 -> M=j+8.
    // Fast path for full tiles (always taken when 16 | nNodes): one predicate,
    // eight unconditional stores. Partial last tile takes the checked path.
    const int rbase = tile * 16 + kh * 8;
    float* dstp = p4 + (size_t)rbase * 4 + m;
    if (tile * 16 + 16 <= nNodes) {
        if (m < 4) {
            #pragma unroll
            for (int j = 0; j < 8; ++j)
                dstp[j * 4] = c[j];
        }
    } else {
        if (m < 4) {
            #pragma unroll
            for (int j = 0; j < 8; ++j)
                if (rbase + j < nNodes) dstp[j * 4] = c[j];
        }
    }
}

// ---------------------------------------------------------------------------
// Phase 2: out[e] = p4[src[e]].xy + p4[dst[e]].zw + b
// (1.6 MB node table is fully L2-resident: 192 MB L2)
// ---------------------------------------------------------------------------
__global__ void __launch_bounds__(256)
edge_mlp_edge_gather(const float* __restrict__ p4,
                     const long long* __restrict__ src,
                     const long long* __restrict__ dst,
                     const float* __restrict__ bias,
                     float* __restrict__ out, int nEdges)
{
    const int e = blockIdx.x * blockDim.x + threadIdx.x;
    if (e >= nEdges) return;
    const long long s = src[e];
    const long long d = dst[e];
    const float2 bb = *(const float2*)bias;
    const float2 p  = *(const float2*)(p4 + s * 4);       // 16B aligned
    const float2 q  = *(const float2*)(p4 + d * 4 + 2);   // 8B aligned
    float2 r;
    r.x = p.x + q.x + bb.x;
    r.y = p.y + q.y + bb.y;
    *(float2*)(out + (size_t)e * 2) = r;
}

extern "C" void kernel_launch(void* const* d_in, const int* in_sizes, int n_in,
                              void* d_out, int out_size, void* d_ws, size_t ws_size,
                              hipStream_t stream) {
    const float*     h   = (const float*)d_in[0];       // (nNodes, 128) f32
    const long long* src = (const long long*)d_in[1];   // (nEdges,) i64
    const long long* dst = (const long long*)d_in[2];   // (nEdges,) i64
    const float*     W   = (const float*)d_in[3];       // (2, 256) f32
    const float*     b   = (const float*)d_in[4];       // (2,) f32
    float*           out = (float*)d_out;               // (nEdges, 2) f32

    const int nNodes = in_sizes[0] / D_DIM;
    const int nEdges = in_sizes[1];

    float* p4 = (float*)d_ws;                           // nNodes * 4 floats (1.6 MB)

    const int nTiles = (nNodes + 15) / 16;
    const int wavesPerBlock = 8;                        // 256 threads = 8 wave32
    dim3 g1((nTiles + wavesPerBlock - 1) / wavesPerBlock);
    edge_mlp_node_proj<<<g1, wavesPerBlock * 32, 0, stream>>>(h, W, p4, nNodes, nTiles);

    dim3 g2((nEdges + 255) / 256);
    edge_mlp_edge_gather<<<g2, 256, 0, stream>>>(p4, src, dst, b, out, nEdges);
}